// bayes_linear_2L_46471546143005
// MI455X (gfx1250) — compile-verified
//
#include <hip/hip_runtime.h>
#include <hip/hip_bf16.h>

// ---------------------------------------------------------------------------
// Bayesian 2-layer MLP forward for MI455X (gfx1250), fp32 via V_WMMA_F32_16X16X4_F32.
// Compute-bound (~70 GFLOP vs ~0.65 GB HBM traffic), so keep full fp32 on the
// matrix pipe. Tiles move global->LDS via GLOBAL_LOAD_ASYNC_TO_LDS_B128 (no VGPR
// staging, no ds_store, ASYNCcnt-tracked) into double-buffered row-major tiles:
//   A[m][k] stride 20 -> A-frag = one ds_load_b64 (conflict-free)
//   B[k][n] stride 144 -> B-frag = ds_load_2addr_b32 (+0/+144; all banks hit once)
// Fragment registers are ping-ponged so DS latency overlaps WMMA issue.
// ---------------------------------------------------------------------------

typedef __attribute__((ext_vector_type(2))) float v2f;
typedef __attribute__((ext_vector_type(8))) float v8f;
typedef int v4i __attribute__((vector_size(4 * sizeof(int))));

#define LOG_SQRT_2PI 0.91893853320467274f

#ifndef __has_builtin
#define __has_builtin(x) 0
#endif
#if __has_builtin(__builtin_amdgcn_global_load_async_to_lds_b128)
#define HAS_ASYNC_LDS 1
#else
#define HAS_ASYNC_LDS 0
#endif

#if HAS_ASYNC_LDS
#define GPTR(p) ((__attribute__((address_space(1))) v4i*)(p))
#define LPTR(p) ((__attribute__((address_space(3))) v4i*)(p))
#if __has_builtin(__builtin_amdgcn_s_wait_asynccnt)
#define WAIT_ASYNC0() __builtin_amdgcn_s_wait_asynccnt(0)
#else
#define WAIT_ASYNC0() asm volatile("s_wait_asynccnt 0x0" ::: "memory")
#endif
#endif

// ---------------- init: zero scalar sums and W3 padding ----------------
__global__ void init_kernel(float* __restrict__ sums, float* __restrict__ W3pad, int n3) {
  int i = blockIdx.x * blockDim.x + threadIdx.x;
  if (i < 2) sums[i] = 0.0f;
  for (int idx = i; idx < n3; idx += gridDim.x * blockDim.x) W3pad[idx] = 0.0f;
}

// ---------------- weight sampling + log-likelihood reduction ----------------
// W = mu + (1e-6 + softplus(p)) * eps ; (W-mu)/std == eps exactly.
__global__ __launch_bounds__(256)
void prep_weights(const float* __restrict__ mu, const float* __restrict__ p,
                  const float* __restrict__ eps, float* __restrict__ W,
                  int n, int ncols, int ldw, float* __restrict__ sums) {
  int i = blockIdx.x * blockDim.x + threadIdx.x;
  float lq = 0.0f, lp = 0.0f;
  for (int idx = i; idx < n; idx += gridDim.x * blockDim.x) {
    float pv  = p[idx];
    float sp  = log1pf(__expf(pv));        // softplus
    float std = 1e-6f + sp;
    float e   = eps[idx];
    float w   = fmaf(std, e, mu[idx]);
    int out   = (ncols == ldw) ? idx : (idx / ncols) * ldw + (idx % ncols);
    W[out] = w;
    lq += -LOG_SQRT_2PI - __logf(std) - 0.5f * e * e;
    lp += -LOG_SQRT_2PI - 0.5f * w * w;
  }
  for (int off = 16; off > 0; off >>= 1) {
    lq += __shfl_down(lq, off, 32);
    lp += __shfl_down(lp, off, 32);
  }
  __shared__ float sq[8], sp2[8];
  int lane = threadIdx.x & 31, wid = threadIdx.x >> 5;
  if (lane == 0) { sq[wid] = lq; sp2[wid] = lp; }
  __syncthreads();
  if (threadIdx.x == 0) {
    float tq = 0.0f, tp = 0.0f;
    for (int w = 0; w < (int)(blockDim.x >> 5); ++w) { tq += sq[w]; tp += sp2[w]; }
    atomicAdd(&sums[0], tq);
    atomicAdd(&sums[1], tp);
  }
}

// ---------------- main fp32 WMMA GEMM: C = [relu](A @ W) ----------------
// Block tile 128(M) x 128(N), BK=16. 8 waves: 2(M) x 4(N); each wave computes a
// 64x32 tile = 4x2 subtiles -> 8 v8f accumulators, 32 WMMA per K-iteration.
constexpr int BM = 128, BN = 128, BK = 16;
constexpr int LDK  = 20;            // A row stride (dwords): b64 frag reads conflict-free
constexpr int LDB  = 144;           // B row stride (dwords): 2addr frag reads hit all banks once
constexpr int ABUF = BM * LDK;      // 2560 dwords
constexpr int BBUF = BK * LDB;      // 2304 dwords

__global__ __launch_bounds__(256, 2)
void gemm_relu_f32_wmma(const float* __restrict__ A, const float* __restrict__ Wt,
                        float* __restrict__ C, int M, int N, int K, int doRelu) {
  __shared__ alignas(16) float As[2 * ABUF];   // ping-pong, row-major [m][k]
  __shared__ alignas(16) float Bs[2 * BBUF];   // ping-pong, row-major [k][n]

  const int tid   = threadIdx.x;
  const int lane  = tid & 31;
  const int wid   = tid >> 5;       // 0..7
  const int waveM = wid & 1;        // 2 waves along M
  const int waveN = wid >> 1;       // 4 waves along N
  const int ln    = lane & 15;
  const int lh    = lane >> 4;

  const int rowBase = blockIdx.y * BM;
  const int colBase = blockIdx.x * BN;
  const float* Aptr = A + (long)rowBase * K;

  v8f acc[4][2];
#pragma unroll
  for (int i = 0; i < 4; ++i)
#pragma unroll
    for (int j = 0; j < 2; ++j)
#pragma unroll
      for (int e = 0; e < 8; ++e) acc[i][j][e] = 0.0f;

  const int m0 = waveM * 64;
  const int n0 = waveN * 32;

  // fragment loads: A-frag = contiguous k-pair (b64); B-frag = two b32, +0/+LDB.
  // The B base index is laundered through an empty asm so the compiler pairs the
  // intra-fragment (+0,+LDB) accesses into one ds_load_2addr_b32 (consecutive
  // VGPRs) instead of cross-fragment pairs that need repacking movs.
  auto loadFrag = [&](int kk, int buf, v2f (&af)[4], v2f (&bf)[2]) {
    const int kr = kk + 2 * lh;                   // this half-wave's K row
    const float* ab = &As[buf * ABUF + kr];
#pragma unroll
    for (int i = 0; i < 4; ++i)
      af[i] = *(const v2f*)(&ab[(m0 + i * 16 + ln) * LDK]);
#pragma unroll
    for (int j = 0; j < 2; ++j) {
      int bidx = buf * BBUF + kr * LDB + n0 + j * 16 + ln;
      asm("" : "+v"(bidx));
      bf[j][0] = Bs[bidx];
      bf[j][1] = Bs[bidx + LDB];
    }
  };

#if HAS_ASYNC_LDS
  // async global->LDS: 2x b128 per thread for A, 2x b128 per thread for B
  auto asyncTile = [&](int k0, int buf) {
#pragma unroll
    for (int s = 0; s < 2; ++s) {
      int l = tid + 256 * s;
      int row = l >> 2, cg = l & 3;               // A: 128 rows x 4 groups of 4 floats
      __builtin_amdgcn_global_load_async_to_lds_b128(
          GPTR(Aptr + (long)row * K + k0 + cg * 4),
          LPTR(&As[buf * ABUF + row * LDK + cg * 4]), 0, 0);
    }
#pragma unroll
    for (int s = 0; s < 2; ++s) {
      int l = tid + 256 * s;
      int k = l >> 5, cg = l & 31;                // B: 16 rows x 32 groups of 4 floats
      __builtin_amdgcn_global_load_async_to_lds_b128(
          GPTR(Wt + (long)(k0 + k) * N + colBase + cg * 4),
          LPTR(&Bs[buf * BBUF + k * LDB + cg * 4]), 0, 0);
    }
  };

  asyncTile(0, 0);
  WAIT_ASYNC0();
  __syncthreads();

  int cur = 0;
  for (int k0 = 0; k0 < K; k0 += BK) {
    const bool has_next = (k0 + BK) < K;
    if (has_next) asyncTile(k0 + BK, cur ^ 1);    // cur^1 unread this iteration

    v2f af[2][4], bf[2][2];
    loadFrag(0, cur, af[0], bf[0]);
#pragma unroll
    for (int kk = 0; kk < BK; kk += 4) {
      const int fc = (kk >> 2) & 1;
      if (kk + 4 < BK) loadFrag(kk + 4, cur, af[fc ^ 1], bf[fc ^ 1]);
#pragma unroll
      for (int i = 0; i < 4; ++i)
#pragma unroll
        for (int j = 0; j < 2; ++j)
          acc[i][j] = __builtin_amdgcn_wmma_f32_16x16x4_f32(
              false, af[fc][i], false, bf[fc][j], (short)0, acc[i][j], false, false);
    }

    WAIT_ASYNC0();                                // my async writes to cur^1 landed
    __syncthreads();                              // everyone's done reading cur
    cur ^= 1;
  }
#else
  // fallback: VGPR-staged tiles (row-major -> plain b128 ds_stores)
  auto loadAB = [&](int k0, float4 (&ar)[2], float4 (&br)[2]) {
    for (int s = 0; s < 2; ++s) {
      int l = tid + 256 * s;
      int row = l >> 2, cg = l & 3;
      ar[s] = *(const float4*)(Aptr + (long)row * K + k0 + cg * 4);
    }
    for (int s = 0; s < 2; ++s) {
      int l = tid + 256 * s;
      int k = l >> 5, cg = l & 31;
      br[s] = *(const float4*)(Wt + (long)(k0 + k) * N + colBase + cg * 4);
    }
  };
  auto storeAB = [&](const float4 (&ar)[2], const float4 (&br)[2], int buf) {
    for (int s = 0; s < 2; ++s) {
      int l = tid + 256 * s;
      int row = l >> 2, cg = l & 3;
      *(float4*)(&As[buf * ABUF + row * LDK + cg * 4]) = ar[s];
    }
    for (int s = 0; s < 2; ++s) {
      int l = tid + 256 * s;
      int k = l >> 5, cg = l & 31;
      *(float4*)(&Bs[buf * BBUF + k * LDB + cg * 4]) = br[s];
    }
  };

  float4 aReg[2], bReg[2];
  loadAB(0, aReg, bReg);
  storeAB(aReg, bReg, 0);
  __syncthreads();

  int cur = 0;
  for (int k0 = 0; k0 < K; k0 += BK) {
    const bool has_next = (k0 + BK) < K;
    float4 aN[2], bN[2];
    if (has_next) loadAB(k0 + BK, aN, bN);

    v2f af[2][4], bf[2][2];
    loadFrag(0, cur, af[0], bf[0]);
#pragma unroll
    for (int kk = 0; kk < BK; kk += 4) {
      const int fc = (kk >> 2) & 1;
      if (kk + 4 < BK) loadFrag(kk + 4, cur, af[fc ^ 1], bf[fc ^ 1]);
#pragma unroll
      for (int i = 0; i < 4; ++i)
#pragma unroll
        for (int j = 0; j < 2; ++j)
          acc[i][j] = __builtin_amdgcn_wmma_f32_16x16x4_f32(
              false, af[fc][i], false, bf[fc][j], (short)0, acc[i][j], false, false);
    }

    if (has_next) { storeAB(aN, bN, cur ^ 1); cur ^= 1; }
    __syncthreads();
  }
#endif

  // epilogue: C/D layout -> VGPR r holds row (r + 8*half), col = lane%16
#pragma unroll
  for (int i = 0; i < 4; ++i) {
    int mBase = rowBase + m0 + i * 16 + 8 * lh;
#pragma unroll
    for (int j = 0; j < 2; ++j) {
      int n = colBase + n0 + j * 16 + ln;
#pragma unroll
      for (int r = 0; r < 8; ++r) {
        float v = acc[i][j][r];
        if (doRelu) v = v > 0.0f ? v : 0.0f;
        C[(long)(mBase + r) * N + n] = v;
      }
    }
  }
}

// ---------------- tail GEMM: y = h2 @ W3 (N=10, padded to 16 in LDS) ------
constexpr int LDA_S = BM + 4;
constexpr int TABUF = BK * LDA_S;

__global__ __launch_bounds__(256, 2)
void gemm_tail_wmma(const float* __restrict__ A, const float* __restrict__ Wt /*K x 16*/,
                    float* __restrict__ Y, int M, int K, int NOUT) {
  __shared__ alignas(16) float Bs[512 * 16];       // entire padded W3: 32 KB
  __shared__ alignas(16) float As[2 * TABUF];      // ping-pong A tiles

  const int tid  = threadIdx.x;
  const int lane = tid & 31;
  const int wid  = tid >> 5;      // wave owns rows wid*16 .. wid*16+15
  const int ln   = lane & 15;
  const int lh   = lane >> 4;

  for (int i = tid; i < (512 * 16) / 4; i += 256)
    *(float4*)(&Bs[i * 4]) = *(const float4*)(Wt + i * 4);

  const int rowBase = blockIdx.x * BM;
  const float* Aptr = A + (long)rowBase * K;

  v8f acc;
#pragma unroll
  for (int e = 0; e < 8; ++e) acc[e] = 0.0f;

  auto loadA = [&](int k0, float4 (&ar)[2]) {
    for (int s = 0; s < 2; ++s) {
      int l = tid + 256 * s;
      int row = l >> 2, cg = l & 3;
      ar[s] = *(const float4*)(Aptr + (long)row * K + k0 + cg * 4);
    }
  };
  auto storeA = [&](const float4 (&ar)[2], int buf) {
    float* base = &As[buf * TABUF];
    for (int s = 0; s < 2; ++s) {
      int l = tid + 256 * s;
      int row = l >> 2, cg = l & 3;
      base[(cg * 4 + 0) * LDA_S + row] = ar[s].x;
      base[(cg * 4 + 1) * LDA_S + row] = ar[s].y;
      base[(cg * 4 + 2) * LDA_S + row] = ar[s].z;
      base[(cg * 4 + 3) * LDA_S + row] = ar[s].w;
    }
  };

  float4 aReg[2];
  loadA(0, aReg);
  storeA(aReg, 0);
  __syncthreads();

  const int m0 = wid * 16;
  int cur = 0;
  for (int k0 = 0; k0 < K; k0 += BK) {
    const bool has_next = (k0 + BK) < K;
    float4 aN[2];
    if (has_next) loadA(k0 + BK, aN);

    const float* ab = &As[cur * TABUF];
#pragma unroll
    for (int kk = 0; kk < BK; kk += 4) {
      v2f afrag, bfrag;
      afrag[0] = ab[(kk + 2 * lh + 0) * LDA_S + m0 + ln];
      afrag[1] = ab[(kk + 2 * lh + 1) * LDA_S + m0 + ln];
      int tb = (k0 + kk + 2 * lh) * 16 + ln;
      asm("" : "+v"(tb));
      bfrag[0] = Bs[tb];
      bfrag[1] = Bs[tb + 16];
      acc = __builtin_amdgcn_wmma_f32_16x16x4_f32(
          false, afrag, false, bfrag, (short)0, acc, false, false);
    }

    if (has_next) { storeA(aN, cur ^ 1); cur ^= 1; }
    __syncthreads();
  }

  const int n = ln;
  if (n < NOUT) {
#pragma unroll
    for (int r = 0; r < 8; ++r) {
      int m = rowBase + wid * 16 + r + 8 * lh;
      Y[(long)m * NOUT + n] = acc[r];
    }
  }
}

// ---------------------------------------------------------------------------
extern "C" void kernel_launch(void* const* d_in, const int* in_sizes, int n_in,
                              void* d_out, int out_size, void* d_ws, size_t ws_size,
                              hipStream_t stream) {
  constexpr int Bsz = 65536, D = 512, DOUT = 10;

  const float* x    = (const float*)d_in[0];
  const float* Wmu1 = (const float*)d_in[1];
  const float* Wp1  = (const float*)d_in[2];
  const float* eps1 = (const float*)d_in[3];
  const float* Wmu2 = (const float*)d_in[4];
  const float* Wp2  = (const float*)d_in[5];
  const float* eps2 = (const float*)d_in[6];
  const float* Wmu3 = (const float*)d_in[7];
  const float* Wp3  = (const float*)d_in[8];
  const float* eps3 = (const float*)d_in[9];

  float* y    = (float*)d_out;                    // 65536*10
  float* sums = y + (long)Bsz * DOUT;             // [tlqw, tlpw]

  float* ws = (float*)d_ws;
  float* W1 = ws;                                 // 512*512
  float* W2 = W1 + D * D;                         // 512*512
  float* W3 = W2 + D * D;                         // 512*16 (padded)
  float* h1 = W3 + D * 16;                        // 65536*512
  float* h2 = h1 + (long)Bsz * D;                 // 65536*512

  init_kernel<<<32, 256, 0, stream>>>(sums, W3, D * 16);
  prep_weights<<<256, 256, 0, stream>>>(Wmu1, Wp1, eps1, W1, D * D, D, D, sums);
  prep_weights<<<256, 256, 0, stream>>>(Wmu2, Wp2, eps2, W2, D * D, D, D, sums);
  prep_weights<<<32, 256, 0, stream>>>(Wmu3, Wp3, eps3, W3, D * DOUT, DOUT, 16, sums);

  dim3 grid(D / BN, Bsz / BM);   // (4, 512); N-tiles fastest -> A reused out of L2
  gemm_relu_f32_wmma<<<grid, 256, 0, stream>>>(x, W1, h1, Bsz, D, D, 1);
  gemm_relu_f32_wmma<<<grid, 256, 0, stream>>>(h1, W2, h2, Bsz, D, D, 1);
  gemm_tail_wmma<<<Bsz / BM, 256, 0, stream>>>(h2, W3, y, Bsz, D, DOUT);
}